// RNN_w_TargetError_44968307589561
// MI455X (gfx1250) — compile-verified
//
#include <hip/hip_runtime.h>
#include <hip/hip_bf16.h>

// ---------------------------------------------------------------------------
// RNN with target-error feedback, MI455X (gfx1250, wave32, WMMA).
//
// Per step t:
//   x += alpha*(-x + bias + r@W_rec + s_t@W_in + err@W_fbk + 0.05*noise_t)
//   r  = relu(x)
//   vel = r @ W_out^T ; cur = vel*dt_s + pos + p_t ; err = cur - tgt_t
//
// Persistent grid: 64 blocks x 256 threads = 512 wave32 waves; each wave owns
// one 16x16 tile of the (128 x 1024) recurrent GEMM via fp32 WMMA
// (v_wmma_f32_16x16x4_f32). All 8 waves of a block share one 16-column slab
// of W_rec, which is staged ONCE into LDS (64 KB, pre-swizzled into K-pairs)
// so every WMMA B-fragment is a single ds_load_b64 with immediate offset.
// Two accumulators overlap the WMMA dependency chains. One device-wide
// barrier per step; readout/error recomputed redundantly per block in LDS.
// ---------------------------------------------------------------------------

#define T_STEPS 500
#define BATCH   128
#define NIN     4
#define NOUT    2
#define NN      1024
#define NBLK    64
#define TPB     256
#define SIGMA_N 0.05f

typedef float v2f __attribute__((ext_vector_type(2)));
typedef float v8f __attribute__((ext_vector_type(8)));

// ---- device-wide sense barrier (state in workspace, re-inited every launch)
__device__ inline void grid_barrier(unsigned* cnt, unsigned* gen) {
    __syncthreads();
    if (threadIdx.x == 0) {
        unsigned g = __hip_atomic_load(gen, __ATOMIC_ACQUIRE, __HIP_MEMORY_SCOPE_AGENT);
        __threadfence();
        unsigned arrived =
            __hip_atomic_fetch_add(cnt, 1u, __ATOMIC_ACQ_REL, __HIP_MEMORY_SCOPE_AGENT);
        if (arrived == NBLK - 1) {
            __hip_atomic_store(cnt, 0u, __ATOMIC_RELAXED, __HIP_MEMORY_SCOPE_AGENT);
            __hip_atomic_fetch_add(gen, 1u, __ATOMIC_ACQ_REL, __HIP_MEMORY_SCOPE_AGENT);
        } else {
            while (__hip_atomic_load(gen, __ATOMIC_ACQUIRE, __HIP_MEMORY_SCOPE_AGENT) == g) {
                __builtin_amdgcn_s_sleep(1);
            }
        }
    }
    __syncthreads();
}

// ---- init: x = hidden_init, r0 = relu(x), zero the barrier counters --------
__global__ void rnn_init_kernel(const float* __restrict__ hidden_init,
                                float* __restrict__ xbuf,
                                float* __restrict__ rbuf,
                                unsigned* __restrict__ bar) {
    int i = blockIdx.x * blockDim.x + threadIdx.x;
    if (i < BATCH * NN) {
        float x = hidden_init[i];
        xbuf[i] = x;
        rbuf[i] = fmaxf(x, 0.0f);   // r buffer 0 (t=0 reads this)
    }
    if (i == 0) { bar[0] = 0u; bar[1] = 0u; }
}

// ---- persistent RNN kernel -------------------------------------------------
__global__ __launch_bounds__(TPB, 1)
void rnn_persistent_kernel(const int*   __restrict__ dt_p,
                           const float* __restrict__ stim,      // (T,B,4)
                           const float* __restrict__ target,    // (T,B,2)
                           const float* __restrict__ perturb,   // (T,B,2)
                           const float* __restrict__ noise,     // (T,B,N)
                           const float* __restrict__ error_init,// (B,2)
                           const float* __restrict__ W_in,      // (4,N)
                           const float* __restrict__ W_rec,     // (N,N)
                           const float* __restrict__ W_fbk,     // (2,N)
                           const float* __restrict__ bias_n,    // (1,N)
                           const float* __restrict__ W_out,     // (2,N)
                           float* __restrict__ cursor_out,      // (T,B,2)
                           float* __restrict__ hidden_out,      // (T,B,N)
                           float* __restrict__ xbuf,            // (B,N)
                           float* __restrict__ rbuf,            // (2,B,N)
                           unsigned* __restrict__ bar) {
    // W_rec column slab for this block, swizzled into (k, k+1) pairs:
    // s_w[(k>>1)*16 + col] = { W_rec[k][n0c+col], W_rec[k+1][n0c+col] }
    __shared__ float2 s_w[(NN / 2) * 16];            // 64 KB
    __shared__ float  s_err[BATCH * NOUT];           // per-block redundant copy
    __shared__ float  s_pos[BATCH * NOUT];

    const int tid = threadIdx.x;
    const int n0c = blockIdx.x << 4;                 // block's 16-column base

    // ---- one-time staging of the W_rec slab (reused for all 500 steps) ----
    for (int i = tid; i < (NN / 2) * 16; i += TPB) {
        const int kk  = (i >> 4) << 1;               // even k
        const int col = i & 15;
        float2 w;
        w.x = W_rec[(size_t)kk * NN + n0c + col];
        w.y = W_rec[(size_t)(kk + 1) * NN + n0c + col];
        s_w[i] = w;
    }
    if (tid < BATCH * NOUT) {
        s_err[tid] = error_init[tid];
        s_pos[tid] = 0.0f;
    }
    __syncthreads();

    const float dtf   = (float)dt_p[0];
    const float alpha = dtf / 10.0f;     // dt / TAU
    const float dt_s  = dtf / 1000.0f;

    const int wave = tid >> 5;                       // 0..7
    const int lane = tid & 31;
    const int m0   = wave << 4;                      // batch-tile base (0..112)
    const int n0   = n0c;                            // neuron-tile base
    const int mrow = lane & 15;                      // M for A, N for B/C/D
    const int half = lane >> 4;                      // K-half selector

    // Per-lane base into the LDS slab: element ((k>>1)+half)*16 + col
    const float2* bfrag = s_w + half * 16 + mrow;

    for (int t = 0; t < T_STEPS; ++t) {
        const float* rprev = rbuf + (size_t)(t & 1) * BATCH * NN;
        float*       rnext = rbuf + (size_t)((t + 1) & 1) * BATCH * NN;

        // ================= phase A: recurrent tile ==========================
        v8f acc0 = {}, acc1 = {};

        // r @ W_rec : K = 1024 in steps of 4; B from LDS, A from L2-resident r
        const float* arow = rprev + (size_t)(m0 + mrow) * NN + 2 * half;
        for (int k = 0; k < NN; k += 8) {
            {
                v2f a, b;
                a.x = arow[k];
                a.y = arow[k + 1];
                float2 w = bfrag[(k >> 1) * 16];     // ds_load_b64, imm offset
                b.x = w.x; b.y = w.y;
                acc0 = __builtin_amdgcn_wmma_f32_16x16x4_f32(
                    false, a, false, b, (short)0, acc0, false, false);
            }
            {
                v2f a, b;
                a.x = arow[k + 4];
                a.y = arow[k + 5];
                float2 w = bfrag[((k + 4) >> 1) * 16];
                b.x = w.x; b.y = w.y;
                acc1 = __builtin_amdgcn_wmma_f32_16x16x4_f32(
                    false, a, false, b, (short)0, acc1, false, false);
            }
        }
        v8f acc = acc0 + acc1;

        // s_t @ W_in : K = 4, exactly one WMMA
        {
            const float* srow = stim + ((size_t)t * BATCH + (m0 + mrow)) * NIN;
            v2f a, b;
            a.x = srow[2 * half];
            a.y = srow[2 * half + 1];
            b.x = W_in[(size_t)(2 * half) * NN + (n0 + mrow)];
            b.y = W_in[(size_t)(2 * half + 1) * NN + (n0 + mrow)];
            acc = __builtin_amdgcn_wmma_f32_16x16x4_f32(
                false, a, false, b, (short)0, acc, false, false);
        }

        // err @ W_fbk : K = 2 padded to 4 (zeros in the half==1 lanes)
        {
            v2f a = {0.0f, 0.0f}, b = {0.0f, 0.0f};
            if (half == 0) {
                a.x = s_err[(m0 + mrow) * NOUT + 0];
                a.y = s_err[(m0 + mrow) * NOUT + 1];
                b.x = W_fbk[n0 + mrow];
                b.y = W_fbk[NN + n0 + mrow];
            }
            acc = __builtin_amdgcn_wmma_f32_16x16x4_f32(
                false, a, false, b, (short)0, acc, false, false);
        }

        // leaky integration + ReLU, write x, r_{t+1}, hidden output
        const size_t tBN = (size_t)t * BATCH * NN;
#pragma unroll
        for (int j = 0; j < 8; ++j) {
            const int bb  = m0 + j + 8 * half;       // C/D layout: M = j + 8*half
            const int nn_ = n0 + mrow;               //             N = lane&15
            const size_t idx = (size_t)bb * NN + nn_;
            float xold = xbuf[idx];
            float nz   = __builtin_nontemporal_load(noise + tBN + idx);
            float pre  = acc[j] + bias_n[nn_] + SIGMA_N * nz;
            float xnew = xold + alpha * (pre - xold);
            float rn   = fmaxf(xnew, 0.0f);
            xbuf[idx]  = xnew;
            rnext[idx] = rn;
            __builtin_nontemporal_store(rn, hidden_out + tBN + idx);
        }

        // ================= device-wide sync (r_{t+1} complete) ==============
        grid_barrier(bar, bar + 1);

        // ================= phase B: redundant readout per block =============
        {
            const int bb = tid >> 1;       // batch 0..127
            const int oo = tid & 1;        // output 0..1
            const float* rrow = rnext + (size_t)bb * NN;
            const float* wrow = W_out + (size_t)oo * NN;
            float v = 0.0f;
            for (int k = 0; k < NN; k += 4) {
                float4 rv = *(const float4*)(rrow + k);
                float4 wv = *(const float4*)(wrow + k);
                v += rv.x * wv.x + rv.y * wv.y + rv.z * wv.z + rv.w * wv.w;
            }
            const size_t oidx = ((size_t)t * BATCH + bb) * NOUT + oo;
            float cur = v * dt_s + s_pos[tid] + perturb[oidx];
            float e   = cur - target[oidx];
            s_err[tid] = e;     // consumed by next phase A (same block only)
            s_pos[tid] = cur;
            if (blockIdx.x == 0) cursor_out[oidx] = cur;
        }
        __syncthreads();
    }
}

extern "C" void kernel_launch(void* const* d_in, const int* in_sizes, int n_in,
                              void* d_out, int out_size, void* d_ws, size_t ws_size,
                              hipStream_t stream) {
    const int*   dt_p        = (const int*)  d_in[0];
    const float* stim        = (const float*)d_in[1];
    const float* target      = (const float*)d_in[2];
    const float* perturb     = (const float*)d_in[3];
    const float* noise       = (const float*)d_in[4];
    const float* hidden_init = (const float*)d_in[5];
    const float* error_init  = (const float*)d_in[6];
    const float* W_in        = (const float*)d_in[7];
    const float* W_rec       = (const float*)d_in[8];
    const float* W_fbk       = (const float*)d_in[9];
    const float* bias_n      = (const float*)d_in[10];
    const float* W_out       = (const float*)d_in[11];

    float* cursor_out = (float*)d_out;                                  // (T,B,2)
    float* hidden_out = (float*)d_out + (size_t)T_STEPS * BATCH * NOUT; // (T,B,N)

    float*    xbuf = (float*)d_ws;                 // 128*1024
    float*    rbuf = xbuf + (size_t)BATCH * NN;    // 2*128*1024 (double buffer)
    unsigned* bar  = (unsigned*)(rbuf + (size_t)2 * BATCH * NN); // 2 counters

    rnn_init_kernel<<<(BATCH * NN + 255) / 256, 256, 0, stream>>>(
        hidden_init, xbuf, rbuf, bar);

    rnn_persistent_kernel<<<NBLK, TPB, 0, stream>>>(
        dt_p, stim, target, perturb, noise, error_init,
        W_in, W_rec, W_fbk, bias_n, W_out,
        cursor_out, hidden_out, xbuf, rbuf, bar);
}